// Linear_6459630814024
// MI455X (gfx1250) — compile-verified
//
#include <hip/hip_runtime.h>
#include <stdint.h>

// ---------------------------------------------------------------------------
// Chebyshev (L-inf) distance: out[b,o] = max_k |x[b,k] - w[o,k]|
// B=1024, OUT=1024, IN=512, fp32.
// VALU-bound problem (max-plus semiring -> WMMA inapplicable). Strategy:
//  - 64x64 output tile per block, 4x4 register tile per thread (256 thr).
//  - K tiled at 32; tiles staged in LDS by the Tensor Data Mover
//    (tensor_load_to_lds), double-buffered, TENSORcnt-synchronized.
//  - TDM pad_enable gives row stride 33 floats -> conflict-free LDS banks.
// ---------------------------------------------------------------------------

#define B_    1024
#define OUT_  1024
#define IN_   512
#define BT    64      // batch rows per block tile
#define OT    64      // out cols per block tile
#define KT    32      // k-tile
#define KP    (KT + 1) // padded LDS row stride (floats); pad written by TDM

typedef uint32_t v4u __attribute__((ext_vector_type(4)));
typedef int32_t  v8i __attribute__((ext_vector_type(8)));
typedef int32_t  v4i __attribute__((ext_vector_type(4)));

// Low 32 bits of a flat shared-aperture address are the LDS byte offset
// (CDNA5 aperture rule).
__device__ __forceinline__ uint32_t lds_offset(const void* p) {
  return (uint32_t)(uintptr_t)p;
}

// Issue one TDM 2D tile load: global (tile_d1 rows x tile_d0 elems, row
// stride stride0 elems, 4-byte elements) -> LDS at lds_addr, with 1 DWORD of
// LDS padding appended every 32 DWORDs (pad_interval=4, pad_amount=0).
__device__ __forceinline__ void tdm_load_tile_2d(
    uint32_t lds_addr, const float* gptr,
    uint32_t tensor_d0, uint32_t tensor_d1, uint64_t stride0_elems,
    uint32_t tile_d0, uint32_t tile_d1)
{
  const uint64_t ga = (uint64_t)(uintptr_t)gptr;

  // ---- D# group 0 (128b): count=1, lds_addr, global_addr[56:0], type=2
  v4u g0;
  g0.x = 0x1u;                                         // count=1, user mode
  g0.y = lds_addr;                                     // LDS byte address
  g0.z = (uint32_t)ga;                                 // global_addr[31:0]
  g0.w = ((uint32_t)(ga >> 32) & 0x01FFFFFFu)          // global_addr[56:32]
         | (2u << 30);                                 // type = 2 ("image")

  // ---- D# group 1 (256b)
  // word0: wg_mask=0 | data_size=2 (4B)<<16 | pad_enable<<20 |
  //        pad_interval=4 (32 DWORDs)<<22 | pad_amount=0 (1 DWORD)<<25
  const uint32_t w0 = (2u << 16) | (1u << 20) | (4u << 22);
  v8i g1;
  g1[0] = (int32_t)w0;
  g1[1] = (int32_t)((tensor_d0 & 0xFFFFu) << 16);                   // bits47:32=0 | dim0 lo16
  g1[2] = (int32_t)((tensor_d0 >> 16) | ((tensor_d1 & 0xFFFFu) << 16)); // dim0 hi16 | dim1 lo16
  g1[3] = (int32_t)((tensor_d1 >> 16) | (tile_d0 << 16));           // dim1 hi16 | tile_dim0
  g1[4] = (int32_t)(tile_d1 & 0xFFFFu);                             // tile_dim1 | tile_dim2=0
  g1[5] = (int32_t)(uint32_t)stride0_elems;                         // dim0_stride[31:0]
  g1[6] = (int32_t)(uint32_t)(stride0_elems >> 32);                 // dim0_stride[47:32] | dim1_stride lo16=0
  g1[7] = 0;                                                        // dim1_stride hi

  // ---- D# groups 2/3: 2D tile; keep tensor_dim2=1 (benign), rest zero.
  v4i g2 = {1, 0, 0, 0};
  v4i g3 = {0, 0, 0, 0};
  v8i g4 = {0, 0, 0, 0, 0, 0, 0, 0};   // 5th operand (clang-23 form), zero

  // amdgpu-toolchain (clang-23) 6-arg form; cpol=0 (RT, device scope default).
  __builtin_amdgcn_tensor_load_to_lds(g0, g1, g2, g3, g4, 0);
}

__global__ __launch_bounds__(256)
void cheb_inf_kernel(const float* __restrict__ x,
                     const float* __restrict__ w,
                     float* __restrict__ out)
{
  __shared__ float xs[2][BT][KP];   // 2 x 64 x 33 floats
  __shared__ float ws[2][OT][KP];

  const int tid  = threadIdx.x;
  const int tx   = tid & 15;        // 16 threads across output cols
  const int ty   = tid >> 4;        // 16 threads across batch rows
  const int row0 = blockIdx.y * BT; // batch-row base
  const int col0 = blockIdx.x * OT; // out-col base

  const int NKT = IN_ / KT;         // 16 k-tiles

  // Prologue: wave 0 DMAs k-tile 0 for both operands. TDM ignores EXEC; the
  // branch keeps waves 1..7 from duplicating the transfer.
  if (tid < 32) {
    tdm_load_tile_2d(lds_offset(&xs[0][0][0]), x + (size_t)row0 * IN_,
                     IN_, B_,  IN_, KT, BT);
    tdm_load_tile_2d(lds_offset(&ws[0][0][0]), w + (size_t)col0 * IN_,
                     IN_, OUT_, IN_, KT, OT);
  }

  float acc[4][4];
#pragma unroll
  for (int i = 0; i < 4; ++i)
#pragma unroll
    for (int j = 0; j < 4; ++j) acc[i][j] = 0.0f;

  for (int t = 0; t < NKT; ++t) {
    const int cur = t & 1;

    if (tid < 32) {
      if (t + 1 < NKT) {
        // Double-buffer: issue tile t+1, then wait until only those 2 TDM
        // ops remain outstanding (TDM is in-order per wave) => tile t ready.
        const int nxt = (t + 1) & 1;
        const int k1  = (t + 1) * KT;
        tdm_load_tile_2d(lds_offset(&xs[nxt][0][0]),
                         x + (size_t)row0 * IN_ + k1, IN_, B_,  IN_, KT, BT);
        tdm_load_tile_2d(lds_offset(&ws[nxt][0][0]),
                         w + (size_t)col0 * IN_ + k1, IN_, OUT_, IN_, KT, OT);
        __builtin_amdgcn_s_wait_tensorcnt(2);
      } else {
        __builtin_amdgcn_s_wait_tensorcnt(0);
      }
    }
    __syncthreads();   // publish tile t to all waves

    // Compute on buffer `cur`. Row stride 33 floats => bank-conflict-free:
    // x reads broadcast across tx; w reads hit 16 distinct banks across tx.
#pragma unroll 4
    for (int k = 0; k < KT; ++k) {
      float xv[4], wv[4];
#pragma unroll
      for (int i = 0; i < 4; ++i) xv[i] = xs[cur][4 * ty + i][k];
#pragma unroll
      for (int j = 0; j < 4; ++j) wv[j] = ws[cur][4 * tx + j][k];
#pragma unroll
      for (int i = 0; i < 4; ++i)
#pragma unroll
        for (int j = 0; j < 4; ++j)
          acc[i][j] = fmaxf(acc[i][j], __builtin_fabsf(xv[i] - wv[j]));
    }
    __syncthreads();   // all reads of buffer `cur` done before TDM overwrites
  }

  // Store 4x4 tile; columns are contiguous -> float4 stores.
#pragma unroll
  for (int i = 0; i < 4; ++i) {
    float4 v = make_float4(acc[i][0], acc[i][1], acc[i][2], acc[i][3]);
    *(float4*)(out + (size_t)(row0 + 4 * ty + i) * OUT_ + col0 + 4 * tx) = v;
  }
}

extern "C" void kernel_launch(void* const* d_in, const int* in_sizes, int n_in,
                              void* d_out, int out_size, void* d_ws, size_t ws_size,
                              hipStream_t stream) {
  const float* x = (const float*)d_in[0];   // inputs  (1024 x 512)
  const float* w = (const float*)d_in[1];   // weights (1024 x 512)
  float* out = (float*)d_out;               // (1024 x 1024)

  dim3 grid(OUT_ / OT, B_ / BT);            // 16 x 16 = 256 blocks
  cheb_inf_kernel<<<grid, 256, 0, stream>>>(x, w, out);
}